// ModelcompressConv_56916906607112
// MI455X (gfx1250) — compile-verified
//
#include <hip/hip_runtime.h>
#include <hip/hip_bf16.h>

typedef __attribute__((ext_vector_type(2))) float v2f;
typedef __attribute__((ext_vector_type(8))) float v8f;

#define N_NODES 100000
#define N_EDGES 1600000
#define D_FEAT  64

// ---------------------------------------------------------------------------
// Bias broadcast via V_WMMA_F32_16X16X4_F32.
//   D(16x16) = A(16x4) x B(4x16),  A[:,0] = 1, B[0,:] = bias[c0..c0+15]
// => D[i][j] = bias[c0+j].  One wave per 16x16 output tile.
// A layout (32-bit A 16x4): lanes 0-15 hold K=0..1 in v[0..1], lanes 16-31 K=2..3.
// B layout mirrors: lanes 0-15 are N=0..15 with K=0 in v0; lanes 16-31 K=2.
// So lanes<16 carry {1,0}/{bias,0}, lanes>=16 carry zeros.
// D layout: VGPR r -> M = r + 8*(lane>=16), N = lane&15.
// ---------------------------------------------------------------------------
__global__ __launch_bounds__(256) void bias_init_wmma(const float* __restrict__ bias,
                                                      float* __restrict__ out) {
  const int lane = threadIdx.x & 31;              // wave32
  const int wave = blockIdx.x * (blockDim.x >> 5) + (threadIdx.x >> 5);
  // 25000 waves exactly: 6250 row tiles x 4 col tiles (100000/16, 64/16)
  const int rowTile = wave >> 2;
  const int colTile = wave & 3;
  const int c0   = colTile * 16;
  const int n    = lane & 15;
  const int half = lane >> 4;

  v2f a;
  a.x = (half == 0) ? 1.0f : 0.0f;
  a.y = 0.0f;
  v2f b;
  b.x = (half == 0) ? bias[c0 + n] : 0.0f;
  b.y = 0.0f;
  v8f c = {};
  // 8 args: (neg_a, A, neg_b, B, c_mod, C, reuse_a, reuse_b)
  v8f d = __builtin_amdgcn_wmma_f32_16x16x4_f32(
      /*neg_a=*/false, a, /*neg_b=*/false, b,
      /*c_mod=*/(short)0, c, /*reuse_a=*/false, /*reuse_b=*/false);

  const int m0 = rowTile * 16 + half * 8;
  float* o = out + (size_t)m0 * D_FEAT + c0 + n;
#pragma unroll
  for (int r = 0; r < 8; ++r) {
    o[(size_t)r * D_FEAT] = d[r];
  }
}

// ---------------------------------------------------------------------------
// Edge-parallel scatter: 16 threads per edge, one float4 of the feature row
// each. Gathers are fully coalesced 128b loads; scatter uses relaxed
// agent-scope non-returning f32 atomics (L2-resident output).
// ---------------------------------------------------------------------------
__global__ __launch_bounds__(256) void edge_scatter(const float* __restrict__ feat,
                                                    const float* __restrict__ weight,
                                                    const int* __restrict__ src,
                                                    const int* __restrict__ dst,
                                                    float* __restrict__ out) {
  const unsigned tid = blockIdx.x * 256u + threadIdx.x;
  const unsigned e = tid >> 4;
  if (e >= N_EDGES) return;
  const unsigned q = tid & 15u;

  const int   s = src[e];
  const int   d = dst[e];
  const float w = weight[e];

  const float4 f = ((const float4*)feat)[(size_t)s * (D_FEAT / 4) + q];

  float* o = out + (size_t)d * D_FEAT + q * 4u;
  __hip_atomic_fetch_add(o + 0, f.x * w, __ATOMIC_RELAXED, __HIP_MEMORY_SCOPE_AGENT);
  __hip_atomic_fetch_add(o + 1, f.y * w, __ATOMIC_RELAXED, __HIP_MEMORY_SCOPE_AGENT);
  __hip_atomic_fetch_add(o + 2, f.z * w, __ATOMIC_RELAXED, __HIP_MEMORY_SCOPE_AGENT);
  __hip_atomic_fetch_add(o + 3, f.w * w, __ATOMIC_RELAXED, __HIP_MEMORY_SCOPE_AGENT);
}

extern "C" void kernel_launch(void* const* d_in, const int* in_sizes, int n_in,
                              void* d_out, int out_size, void* d_ws, size_t ws_size,
                              hipStream_t stream) {
  const float* feat   = (const float*)d_in[0];   // [N_NODES, 64]
  const float* weight = (const float*)d_in[1];   // [N_EDGES, 1]
  const float* bias   = (const float*)d_in[2];   // [64]
  const int*   src    = (const int*)d_in[3];     // [N_EDGES]
  const int*   dst    = (const int*)d_in[4];     // [N_EDGES]
  float*       out    = (float*)d_out;           // [N_NODES, 64]

  // 25000 waves = 3125 blocks x 8 waves; exact tiling of 100000x64.
  bias_init_wmma<<<3125, 256, 0, stream>>>(bias, out);

  // 1.6M edges x 16 threads = 25.6M threads = 100000 blocks of 256.
  edge_scatter<<<(N_EDGES * 16) / 256, 256, 0, stream>>>(feat, weight, src, dst, out);
}